// TorchBlackHatModule_4157528343281
// MI455X (gfx1250) — compile-verified
//
#include <hip/hip_runtime.h>
#include <hip/hip_bf16.h>
#include <stdint.h>

#define AS1 __attribute__((address_space(1)))
#define AS3 __attribute__((address_space(3)))

// Problem constants (reference is fixed-shape).
constexpr int B = 16, C = 3, H = 1024, W = 1024;
constexpr int KS = 15, R = KS / 2;          // 15x15 window, 7-px halo
constexpr int TILE = 64;                    // 64x64 output tile
constexpr int BT = 256;                     // 8 wave32 per block
constexpr int IN_W = TILE + 2 * R;          // 78
constexpr int S_IN_STRIDE = IN_W + 2;       // 80 floats -> bank-friendly

// ---------------------------------------------------------------------------
// K1: gray = mean over 3 channels, float4-vectorized (pure bandwidth).
// ---------------------------------------------------------------------------
__global__ __launch_bounds__(256) void gray_kernel(const float* __restrict__ x,
                                                   float* __restrict__ g) {
    const size_t QPP = (size_t)H * W / 4;       // float4s per plane = 2^18
    size_t p4 = (size_t)blockIdx.x * blockDim.x + threadIdx.x;  // < B*QPP
    size_t b  = p4 >> 18;
    size_t i4 = p4 & (QPP - 1);
    const float4* base = (const float4*)x + b * 3 * QPP;
    float4 c0 = base[i4];
    float4 c1 = base[QPP + i4];
    float4 c2 = base[2 * QPP + i4];
    float4 r;
    const float k = 1.0f / 3.0f;
    r.x = (c0.x + c1.x + c2.x) * k;
    r.y = (c0.y + c1.y + c2.y) * k;
    r.z = (c0.z + c1.z + c2.z) * k;
    r.w = (c0.w + c1.w + c2.w) * k;
    ((float4*)g)[p4] = r;
}

// ---------------------------------------------------------------------------
// K2/K3: separable 15x15 max (IS_MAX) or min (!IS_MAX) with edge-replicate
// padding via clamped addresses. Tile loads use CDNA5 async global->LDS
// copies (ASYNCcnt) — no VGPR round trip. Optional pointwise subtraction of
// `subsrc` (for black_hat = closing - gray); subsrc is read at exactly the
// written index per thread, so dst==subsrc aliasing is safe.
// ---------------------------------------------------------------------------
template <bool IS_MAX, bool SUB>
__global__ __launch_bounds__(BT) void morph15_kernel(
    const float* __restrict__ src, const float* __restrict__ subsrc,
    float* __restrict__ dst) {
    __shared__ float s_in[IN_W * S_IN_STRIDE];   // 78 x 80 floats (~24.4 KB)
    __shared__ float s_row[IN_W * TILE];         // 78 x 64 floats (~19.5 KB)

    const int tid = threadIdx.x;
    const int tx0 = blockIdx.x * TILE;
    const int ty0 = blockIdx.y * TILE;
    const int b   = blockIdx.z;
    const size_t plane = (size_t)H * W;
    const float* sp = src + (size_t)b * plane;

    // --- async load 78x78 haloed tile into LDS (clamped = replicate pad) ---
    for (int i = tid; i < IN_W * IN_W; i += BT) {
        int ly = i / IN_W;
        int lx = i - ly * IN_W;
        int gy = ty0 + ly - R;
        gy = gy < 0 ? 0 : (gy > H - 1 ? H - 1 : gy);
        int gx = tx0 + lx - R;
        gx = gx < 0 ? 0 : (gx > W - 1 ? W - 1 : gx);
        const float* gp = sp + (size_t)gy * W + gx;
        float* lp = &s_in[ly * S_IN_STRIDE + lx];
        __builtin_amdgcn_global_load_async_to_lds_b32(
            (AS1 int*)gp, (AS3 int*)lp, /*offset=*/0, /*cpol=*/0);
    }
    __builtin_amdgcn_s_wait_asynccnt(0);
    __syncthreads();

    // --- row pass: 15-tap horizontal extreme for all 78 rows x 64 cols ---
    for (int i = tid; i < IN_W * TILE; i += BT) {
        int ly = i >> 6;            // row 0..77
        int ox = i & (TILE - 1);    // col 0..63
        const float* r = &s_in[ly * S_IN_STRIDE + ox];
        float m = r[0];
#pragma unroll
        for (int k = 1; k < KS; ++k)
            m = IS_MAX ? fmaxf(m, r[k]) : fminf(m, r[k]);
        s_row[ly * TILE + ox] = m;
    }
    __syncthreads();

    // --- col pass: 15-tap vertical extreme, then (optional) subtract+store ---
    for (int i = tid; i < TILE * TILE; i += BT) {
        int oy = i >> 6;
        int ox = i & (TILE - 1);
        const float* col = &s_row[oy * TILE + ox];
        float m = col[0];
#pragma unroll
        for (int k = 1; k < KS; ++k)
            m = IS_MAX ? fmaxf(m, col[k * TILE]) : fminf(m, col[k * TILE]);
        size_t o = (size_t)b * plane + (size_t)(ty0 + oy) * W + (tx0 + ox);
        float v = m;
        if (SUB) v -= subsrc[o];
        dst[o] = v;
    }
}

// ---------------------------------------------------------------------------
// Launch: K1 gray->d_out, K2 dilate(d_out)->ws, K3 erode(ws)-gray -> d_out.
// d_ws needs 64 MiB (dilated plane stack). All on `stream`; no sync calls.
// ---------------------------------------------------------------------------
extern "C" void kernel_launch(void* const* d_in, const int* in_sizes, int n_in,
                              void* d_out, int out_size, void* d_ws,
                              size_t ws_size, hipStream_t stream) {
    (void)in_sizes; (void)n_in; (void)out_size; (void)ws_size;
    const float* x = (const float*)d_in[0];
    float* out = (float*)d_out;   // holds gray, then overwritten with black_hat
    float* dil = (float*)d_ws;    // 16*1024*1024 floats = 64 MiB

    const size_t nq = (size_t)B * H * W / 4;                  // float4 count
    gray_kernel<<<dim3((unsigned)(nq / 256)), dim3(256), 0, stream>>>(x, out);

    dim3 g(W / TILE, H / TILE, B);
    morph15_kernel<true, false><<<g, dim3(BT), 0, stream>>>(out, nullptr, dil);
    morph15_kernel<false, true><<<g, dim3(BT), 0, stream>>>(dil, out, out);
}